// S4Layer_17420387352579
// MI455X (gfx1250) — compile-verified
//
#include <hip/hip_runtime.h>
#include <stdint.h>

// ---------------------------------------------------------------------------
// S4 layer on MI455X (gfx1250):
//   kernel 1: Cauchy generating-function evaluation -> K[L] (complex, in d_ws)
//   kernel 2: per-batch-row fused FFT(16384) -> K-multiply -> IFFT -> real
//             * entire signal lives in LDS (2 x 128 KB ping-pong, 320KB WGP LDS)
//             * radix-16 stages done as complex 16x16 matmuls on
//               V_WMMA_F32_16X16X4_F32 (4 chained K=4 WMMAs per real matmul)
//             * input row DMA'd to LDS via TDM tensor_load_to_lds
// ---------------------------------------------------------------------------

typedef float    v2f  __attribute__((ext_vector_type(2)));
typedef float    v8f  __attribute__((ext_vector_type(8)));
typedef uint32_t u32x4 __attribute__((ext_vector_type(4)));
typedef int32_t  i32x4 __attribute__((ext_vector_type(4)));
typedef int32_t  i32x8 __attribute__((ext_vector_type(8)));

#define L_FFT    16384
#define N_STATE  64
#define NTHREADS 256     // 8 wave32 waves
#define NWAVES   (NTHREADS / 32)
#define TWO_PI   6.283185307179586f

#if defined(__HIP_DEVICE_COMPILE__) && __has_builtin(__builtin_amdgcn_tensor_load_to_lds)
#define USE_TDM 1
#else
#define USE_TDM 0
#endif

__device__ __forceinline__ float2 cmul(float2 a, float2 b) {
  return make_float2(a.x * b.x - a.y * b.y, a.x * b.y + a.y * b.x);
}
__device__ __forceinline__ float2 cadd(float2 a, float2 b) {
  return make_float2(a.x + b.x, a.y + b.y);
}
__device__ __forceinline__ float2 csub(float2 a, float2 b) {
  return make_float2(a.x - b.x, a.y - b.y);
}

// ---------------------------------------------------------------------------
// Kernel 1: S4 DPLR generating function at the L roots of unity.
// One thread per frequency; N=64 inner loop of complex reciprocals + 4 dots.
// ---------------------------------------------------------------------------
__global__ __launch_bounds__(256) void s4_cauchy_kernel(
    const float* __restrict__ Lr, const float* __restrict__ Li,
    const float* __restrict__ Pr, const float* __restrict__ Pi,
    const float* __restrict__ Br, const float* __restrict__ Bi,
    const float* __restrict__ Cr, const float* __restrict__ Ci,
    const float* __restrict__ step_p, float2* __restrict__ Kout) {
  const int l = blockIdx.x * blockDim.x + threadIdx.x;
  if (l >= L_FFT) return;
  const float step = step_p[0];

  float sw, cw;
  __sincosf(-TWO_PI * (float)l / (float)L_FFT, &sw, &cw);
  const float2 om   = make_float2(cw, sw);             // Omega
  const float2 onep = make_float2(1.f + om.x,  om.y);  // 1 + Omega
  const float2 onem = make_float2(1.f - om.x, -om.y);  // 1 - Omega
  const float  idn  = 1.0f / (onep.x * onep.x + onep.y * onep.y);
  const float  gs   = 2.0f / step;
  // g = (2/step) * (1-Om)/(1+Om)
  const float2 g = make_float2(gs * (onem.x * onep.x + onem.y * onep.y) * idn,
                               gs * (onem.y * onep.x - onem.x * onep.y) * idn);
  // c = 2/(1+Om)
  const float2 cfac = make_float2(2.0f * onep.x * idn, -2.0f * onep.y * idn);

  float2 k00 = {0.f, 0.f}, k01 = {0.f, 0.f}, k10 = {0.f, 0.f}, k11 = {0.f, 0.f};
  for (int n = 0; n < N_STATE; ++n) {
    const float2 d   = make_float2(g.x - Lr[n], g.y - Li[n]);
    const float  di  = 1.0f / (d.x * d.x + d.y * d.y);
    const float2 inv = make_float2(d.x * di, -d.y * di);
    const float2 Bn = make_float2(Br[n], Bi[n]);
    const float2 Pn = make_float2(Pr[n], Pi[n]);
    const float2 a0 = make_float2(Cr[n], -Ci[n]);  // conj(C)
    const float2 a1 = make_float2(Pr[n], -Pi[n]);  // conj(P)
    k00 = cadd(k00, cmul(inv, cmul(a0, Bn)));
    k01 = cadd(k01, cmul(inv, cmul(a0, Pn)));
    k10 = cadd(k10, cmul(inv, cmul(a1, Bn)));
    k11 = cadd(k11, cmul(inv, cmul(a1, Pn)));
  }
  const float2 o  = make_float2(1.f + k11.x, k11.y);
  const float  oi = 1.0f / (o.x * o.x + o.y * o.y);
  const float2 r  = make_float2(o.x * oi, -o.y * oi);     // 1/(1+k11)
  Kout[l] = cmul(cfac, csub(k00, cmul(cmul(k01, r), k10)));
}

// ---------------------------------------------------------------------------
// One radix-16 Stockham DIT stage, DFT-16 done on V_WMMA_F32_16X16X4_F32.
//   Y = W16 * diag(twiddle) * X gathered at stride L/16, scattered at stride Ns
// Complex matmul via 3 A-fragment sets: C=cos, SA, SB (= +/- sin per direction)
//   Yr accumulates C*Xr + SA*Xi ; Yi accumulates C*Xi + SB*Xr
// Fragment layout (per documented f32 16x4 A / 16x16 C layouts, mirrored for B):
//   A chunk c: lane m=lane&15, a[e] = W[m][4c + 2*(lane>=16) + e]
//   B chunk c: lane n=lane&15, b[e] = X[4c + 2*(lane>=16) + e][n]
//   D: reg r holds (M = r + 8*(lane>=16), N = lane&15)
// ---------------------------------------------------------------------------
__device__ __forceinline__ void radix16_wmma_stage(
    const float2* __restrict__ src, float2* __restrict__ dst,
    int Ns, int logNs, float sgn,
    const v2f* Cf, const v2f* SA, const v2f* SB, int lane, int wv) {
  const int n  = lane & 15;
  const int kh = (lane >> 4) << 1;
  const int Lr = L_FFT / 16;  // 1024 columns
  const float tw = sgn * TWO_PI / (float)(16 * Ns);

  for (int tile = wv; tile < Lr / 16; tile += NWAVES) {
    const int j = (tile << 4) + n;  // this lane's column
    const int q = j & (Ns - 1);
    const int s = j >> logNs;

    // Gather + twiddle the B fragments (real & imag).
    v2f br[4], bi[4];
#pragma unroll
    for (int c = 0; c < 4; ++c) {
#pragma unroll
      for (int e = 0; e < 2; ++e) {
        const int p = 4 * c + kh + e;
        const float2 v = src[j + p * Lr];
        float swv, cwv;
        __sincosf(tw * (float)(q * p), &swv, &cwv);
        br[c][e] = v.x * cwv - v.y * swv;
        bi[c][e] = v.x * swv + v.y * cwv;
      }
    }

    v8f yr = {0.f, 0.f, 0.f, 0.f, 0.f, 0.f, 0.f, 0.f};
    v8f yi = {0.f, 0.f, 0.f, 0.f, 0.f, 0.f, 0.f, 0.f};
#pragma unroll
    for (int c = 0; c < 4; ++c) {
      yr = __builtin_amdgcn_wmma_f32_16x16x4_f32(false, Cf[c], false, br[c],
                                                 (short)0, yr, false, false);
      yi = __builtin_amdgcn_wmma_f32_16x16x4_f32(false, Cf[c], false, bi[c],
                                                 (short)0, yi, false, false);
    }
#pragma unroll
    for (int c = 0; c < 4; ++c) {
      yr = __builtin_amdgcn_wmma_f32_16x16x4_f32(false, SA[c], false, bi[c],
                                                 (short)0, yr, false, false);
      yi = __builtin_amdgcn_wmma_f32_16x16x4_f32(false, SB[c], false, br[c],
                                                 (short)0, yi, false, false);
    }

    // Stockham scatter: dst[q + s*16*Ns + k*Ns]
    const int ob = q + (s << (logNs + 4));
#pragma unroll
    for (int r = 0; r < 8; ++r) {
      const int k = r + ((lane >> 4) << 3);
      dst[ob + (k << logNs)] = make_float2(yr[r], yi[r]);
    }
  }
}

// Final radix-4 stage (Ns = 4096) in plain VALU (DFT-4 is adds/swaps).
__device__ __forceinline__ void radix4_stage(const float2* __restrict__ src,
                                             float2* __restrict__ dst,
                                             float sgn, int tid) {
  const int Ns = 4096;
  const int Lr = L_FFT / 4;  // 4096 columns; s==0, q==j
  const float tw = sgn * TWO_PI / (float)(4 * Ns);
  for (int j = tid; j < Lr; j += NTHREADS) {
    float2 t0 = src[j];
    float2 t1 = src[j + Lr];
    float2 t2 = src[j + 2 * Lr];
    float2 t3 = src[j + 3 * Lr];
    float s1, c1, s2, c2, s3, c3;
    __sincosf(tw * (float)j, &s1, &c1);
    __sincosf(tw * (float)(2 * j), &s2, &c2);
    __sincosf(tw * (float)(3 * j), &s3, &c3);
    t1 = cmul(t1, make_float2(c1, s1));
    t2 = cmul(t2, make_float2(c2, s2));
    t3 = cmul(t3, make_float2(c3, s3));
    const float2 s02 = cadd(t0, t2), d02 = csub(t0, t2);
    const float2 s13 = cadd(t1, t3), d13 = csub(t1, t3);
    const float2 id13 = make_float2(-sgn * d13.y, sgn * d13.x);  // sgn*i*d13
    dst[j]          = cadd(s02, s13);
    dst[j + Ns]     = cadd(d02, id13);
    dst[j + 2 * Ns] = csub(s02, s13);
    dst[j + 3 * Ns] = csub(d02, id13);
  }
}

// ---------------------------------------------------------------------------
// Kernel 2: fused FFT -> K multiply -> IFFT -> real, one block per batch row.
// Dynamic LDS: 2 * 16384 complex floats = 256 KB ping-pong.
// ---------------------------------------------------------------------------
__global__ __launch_bounds__(NTHREADS, 1) void s4_fftconv_kernel(
    const float* __restrict__ x, const float2* __restrict__ Kf,
    float* __restrict__ out) {
  extern __shared__ float2 smem[];
  float2* bufA = smem;
  float2* bufB = smem + L_FFT;

  const int tid  = threadIdx.x;
  const int lane = tid & 31;
  const int wv   = tid >> 5;
  const int bid  = blockIdx.x;

  // --- Precompute DFT-16 A fragments: cos, +sin, -sin -------------------
  const int m  = lane & 15;
  const int kh = (lane >> 4) << 1;
  v2f Cf[4], Sf[4], Sn[4];
#pragma unroll
  for (int c = 0; c < 4; ++c) {
#pragma unroll
    for (int e = 0; e < 2; ++e) {
      const int p = 4 * c + kh + e;
      float sv, cv;
      __sincosf(TWO_PI * (float)(m * p) * (1.0f / 16.0f), &sv, &cv);
      Cf[c][e] = cv;
      Sf[c][e] = sv;
      Sn[c][e] = -sv;
    }
  }

  // --- Load input row (64 KB) into LDS --------------------------------
  const float* xrow = x + (size_t)bid * L_FFT;
#if USE_TDM
  if (wv == 0) {
    // 1D TDM descriptor: 16384 x f32, global -> LDS staging (bufB region).
    const uint64_t ga  = (uint64_t)(uintptr_t)xrow;
    const uint32_t lds = (uint32_t)(uintptr_t)bufB;  // low 32b = LDS byte offset
    u32x4 g0;
    g0[0] = 1u;                                  // count=1, user descriptor
    g0[1] = lds;                                 // lds_addr
    g0[2] = (uint32_t)ga;                        // global_addr[31:0]
    g0[3] = ((uint32_t)(ga >> 32) & 0x01FFFFFFu) // global_addr[56:32]
            | (2u << 30);                        // type = 2 ("image")
    i32x8 g1;
    g1[0] = (int)(2u << 16);      // workgroup_mask=0, data_size=4B
    g1[1] = (int)(16384u << 16);  // tensor_dim0 = 16384 (low 16)
    g1[2] = (int)(1u << 16);      // tensor_dim0 hi16=0 | tensor_dim1 = 1
    g1[3] = (int)(16384u << 16);  // tensor_dim1 hi16=0 | tile_dim0 = 16384
    g1[4] = 1;                    // tile_dim1 = 1, tile_dim2 = 0
    g1[5] = 16384;                // tensor_dim0_stride (low 32)
    g1[6] = 0;
    g1[7] = 0;
    i32x4 g2 = {0, 0, 0, 0};
    i32x4 g3 = {0, 0, 0, 0};
#if __clang_major__ >= 23
    i32x8 g4 = {0, 0, 0, 0, 0, 0, 0, 0};
    __builtin_amdgcn_tensor_load_to_lds(g0, g1, g2, g3, g4, 0);
#else
    __builtin_amdgcn_tensor_load_to_lds(g0, g1, g2, g3, 0);
#endif
    __builtin_amdgcn_s_wait_tensorcnt(0);
  }
  __syncthreads();
  const float* staged = (const float*)bufB;
  for (int i = tid; i < L_FFT; i += NTHREADS)
    bufA[i] = make_float2(staged[i], 0.0f);
  __syncthreads();
#else
  for (int i = tid; i < L_FFT; i += NTHREADS)
    bufA[i] = make_float2(xrow[i], 0.0f);
  __syncthreads();
#endif

  // --- Forward FFT: radix 16,16,16,4 (Stockham, natural order) --------
  radix16_wmma_stage(bufA, bufB, 1, 0, -1.f, Cf, Sf, Sn, lane, wv);
  __syncthreads();
  radix16_wmma_stage(bufB, bufA, 16, 4, -1.f, Cf, Sf, Sn, lane, wv);
  __syncthreads();
  radix16_wmma_stage(bufA, bufB, 256, 8, -1.f, Cf, Sf, Sn, lane, wv);
  __syncthreads();
  radix4_stage(bufB, bufA, -1.f, tid);
  __syncthreads();

  // --- Pointwise multiply by S4 kernel (fold in 1/L for the inverse) ---
  const float scale = 1.0f / (float)L_FFT;
  for (int i = tid; i < L_FFT; i += NTHREADS) {
    const float2 p = cmul(bufA[i], Kf[i]);
    bufA[i] = make_float2(p.x * scale, p.y * scale);
  }
  __syncthreads();

  // --- Inverse FFT (conjugate twiddles / matrices) ---------------------
  radix16_wmma_stage(bufA, bufB, 1, 0, +1.f, Cf, Sn, Sf, lane, wv);
  __syncthreads();
  radix16_wmma_stage(bufB, bufA, 16, 4, +1.f, Cf, Sn, Sf, lane, wv);
  __syncthreads();
  radix16_wmma_stage(bufA, bufB, 256, 8, +1.f, Cf, Sn, Sf, lane, wv);
  __syncthreads();
  radix4_stage(bufB, bufA, +1.f, tid);
  __syncthreads();

  // --- Write real part -------------------------------------------------
  float* orow = out + (size_t)bid * L_FFT;
  for (int i = tid; i < L_FFT; i += NTHREADS) orow[i] = bufA[i].x;
}

// ---------------------------------------------------------------------------
extern "C" void kernel_launch(void* const* d_in, const int* in_sizes, int n_in,
                              void* d_out, int out_size, void* d_ws,
                              size_t ws_size, hipStream_t stream) {
  (void)n_in; (void)out_size; (void)ws_size;
  const float* x  = (const float*)d_in[0];
  const float* Lr = (const float*)d_in[1];
  const float* Li = (const float*)d_in[2];
  const float* Pr = (const float*)d_in[3];
  const float* Pi = (const float*)d_in[4];
  const float* Br = (const float*)d_in[5];
  const float* Bi = (const float*)d_in[6];
  const float* Cr = (const float*)d_in[7];
  const float* Ci = (const float*)d_in[8];
  const float* st = (const float*)d_in[9];

  float2* Kbuf = (float2*)d_ws;          // 16384 * 8 B = 128 KB scratch
  float*  outp = (float*)d_out;

  s4_cauchy_kernel<<<L_FFT / 256, 256, 0, stream>>>(Lr, Li, Pr, Pi, Br, Bi,
                                                    Cr, Ci, st, Kbuf);

  const int batch = in_sizes[0] / L_FFT;
  s4_fftconv_kernel<<<batch, NTHREADS, 2 * L_FFT * sizeof(float2), stream>>>(
      x, Kbuf, outp);
}